// RoiAlian_11450382811678
// MI455X (gfx1250) — compile-verified
//
#include <hip/hip_runtime.h>
#include <hip/hip_bf16.h>

typedef float v2f __attribute__((ext_vector_type(2)));
typedef float v8f __attribute__((ext_vector_type(8)));

#define POOL_H 7
#define POOL_W 7
#define IMG_H  100
#define IMG_W  100
#define CH     256

// One block = (roi, pooled column pw). 256 threads = 8 wave32.
// Each wave owns 32 channels as two 16-channel WMMA N-tiles sharing one A.
// out[ph, c] = sum_{k'=(j, sy, tap)} A[ph,k'] * B[k',c]  (separable bilinear + 2x2 pool)
__global__ __launch_bounds__(256) void roi_align_wmma_kernel(
    const float* __restrict__ x,      // (1, 100, 100, 256) NHWC
    const float* __restrict__ rois,   // (N, 4)
    float* __restrict__ out,          // (N, 7, 7, 256)
    int n_rois)
{
    // Precombined per-ROI tables (weights include validity; x weights include 0.25 pool)
    __shared__ float sAy0[16], sAy1[16];         // (1-wy)*vy , wy*vy
    __shared__ float sBx0[16], sBx1[16];         // (1-wx)*vx*0.25 , wx*vx*0.25
    __shared__ int   sRow0[16], sRow1[16];       // row * (W*C)  element offsets
    __shared__ int   sCol0[16], sCol1[16];       // col * C      element offsets

    const int roi = blockIdx.x;
    const int pw  = blockIdx.y;
    const int tid = threadIdx.x;
    if (roi >= n_rois) return;

    // ---------------- per-ROI scalar setup ----------------
    const float* rp = rois + (size_t)roi * 4;
    const float x0 = rp[0] + 1.0f, y0 = rp[1] + 1.0f;
    const float x1 = rp[2] + 1.0f, y1 = rp[3] + 1.0f;
    const float w  = x1 - x0, h = y1 - y0;
    // bx1 = (x0 + w/14 - 0.5)/100 ; grid over (Wp-1)=101, 14 points, step /13
    const float bx1   = (x0 + w * (1.0f / 14.0f) - 0.5f) * 0.01f;
    const float by1   = (y0 + h * (1.0f / 14.0f) - 0.5f) * 0.01f;
    const float stepx = (w * 0.01f) * (101.0f / 13.0f);
    const float stepy = (h * 0.01f) * (101.0f / 13.0f);

    if (tid < 14) {
        // y-axis terms for sample row sy = tid (padded image Hp = 102)
        const float c   = by1 * 101.0f + (float)tid * stepy;
        const float lo  = floorf(c);
        const float wyf = c - lo;
        const float vy  = (c >= 0.0f && c <= 101.0f) ? 1.0f : 0.0f;
        sAy0[tid] = (1.0f - wyf) * vy;
        sAy1[tid] = wyf * vy;
        const int loi = (int)fminf(fmaxf(lo, 0.0f), 101.0f);
        const int hii = (int)fminf(fmaxf(lo + 1.0f, 0.0f), 101.0f);
        // symmetric pad width 1: xp[i] == x[clamp(i-1, 0, H-1)]
        sRow0[tid] = min(max(loi - 1, 0), IMG_H - 1) * (IMG_W * CH);
        sRow1[tid] = min(max(hii - 1, 0), IMG_H - 1) * (IMG_W * CH);
    } else if (tid >= 32 && tid < 46) {
        const int t = tid - 32;
        const float c   = bx1 * 101.0f + (float)t * stepx;
        const float lo  = floorf(c);
        const float wxf = c - lo;
        const float vx  = (c >= 0.0f && c <= 101.0f) ? 1.0f : 0.0f;
        sBx0[t] = (1.0f - wxf) * vx * 0.25f;
        sBx1[t] = wxf * vx * 0.25f;
        const int loi = (int)fminf(fmaxf(lo, 0.0f), 101.0f);
        const int hii = (int)fminf(fmaxf(lo + 1.0f, 0.0f), 101.0f);
        sCol0[t] = min(max(loi - 1, 0), IMG_W - 1) * CH;
        sCol1[t] = min(max(hii - 1, 0), IMG_W - 1) * CH;
    }
    __syncthreads();

    const int lane = tid & 31;
    const int half = lane >> 4;          // A/B: selects K pair; C/D: selects M+8
    const int ln   = lane & 15;          // A: M (=ph); B/C/D: N (=channel-in-tile)
    const unsigned choff = (unsigned)((tid >> 5) * 32 + ln);  // lane's channel

    // x-direction tap columns + pooled weights for this pw
    float bxw[4];
    unsigned colo[4];
    {
        const int sx0 = 2 * pw, sx1 = 2 * pw + 1;
        bxw[0] = sBx0[sx0]; bxw[1] = sBx1[sx0];
        bxw[2] = sBx0[sx1]; bxw[3] = sBx1[sx1];
        colo[0] = (unsigned)sCol0[sx0] + choff;
        colo[1] = (unsigned)sCol1[sx0] + choff;
        colo[2] = (unsigned)sCol0[sx1] + choff;
        colo[3] = (unsigned)sCol1[sx1] + choff;
    }

    // Hoist this lane's per-r y terms: sy = 2r + half
    float    a0[7], a1[7];
    unsigned ro0[7], ro1[7];
#pragma unroll
    for (int r = 0; r < 7; ++r) {
        const int sy = 2 * r + half;
        a0[r]  = sAy0[sy];
        a1[r]  = sAy1[sy];
        ro0[r] = (unsigned)sRow0[sy];
        ro1[r] = (unsigned)sRow1[sy];
    }

    v8f accA = {};   // channels [cbase, cbase+16)
    v8f accB = {};   // channels [cbase+16, cbase+32)

#pragma unroll
    for (int j = 0; j < 4; ++j) {
        const float    bw = bxw[j];
        const unsigned co = colo[j];
#pragma unroll
        for (int r = 0; r < 7; ++r) {
            // lane's K pair: K = 2*half + v (v=0,1) -> sy = 2r + half, tap = v
            const bool hit = (ln == r);
            v2f A;
            A.x = hit ? bw * a0[r] : 0.0f;   // tap = lo
            A.y = hit ? bw * a1[r] : 0.0f;   // tap = hi
            const float* p0 = x + (size_t)(ro0[r] + co);   // 32-bit offset, saddr form
            const float* p1 = x + (size_t)(ro1[r] + co);
            v2f B0, B1;
            B0.x = p0[0];  B0.y = p1[0];     // K=2h (lo row), K=2h+1 (hi row)
            B1.x = p0[16]; B1.y = p1[16];
            accA = __builtin_amdgcn_wmma_f32_16x16x4_f32(
                       false, A, false, B0, (short)0, accA, false, false);
            accB = __builtin_amdgcn_wmma_f32_16x16x4_f32(
                       false, A, false, B1, (short)0, accB, false, false);
        }
    }

    // D layout: VGPR d, lanes 0-15 -> M=d (=ph), N=ln; lanes 16-31 -> M=d+8 (ph>6, unused)
    if (half == 0) {
        const unsigned cb = (unsigned)((tid >> 5) * 32) + (unsigned)ln;
#pragma unroll
        for (int d = 0; d < 7; ++d) {
            const size_t o = (((size_t)roi * POOL_H + d) * POOL_W + pw) * CH + cb;
            out[o]      = accA[d];
            out[o + 16] = accB[d];
        }
    }
}

extern "C" void kernel_launch(void* const* d_in, const int* in_sizes, int n_in,
                              void* d_out, int out_size, void* d_ws, size_t ws_size,
                              hipStream_t stream) {
    const float* x    = (const float*)d_in[0];   // (1,100,100,256) f32
    const float* rois = (const float*)d_in[1];   // (N,4) f32
    float* out = (float*)d_out;                  // (N,7,7,256) f32
    const int n_rois = in_sizes[1] / 4;

    dim3 grid(n_rois, POOL_W);
    roi_align_wmma_kernel<<<grid, 256, 0, stream>>>(x, rois, out, n_rois);
}